// AttnHead_1632087573143
// MI455X (gfx1250) — compile-verified
//
#include <hip/hip_runtime.h>

// CDNA5 (gfx1250) fp32 WMMA attention-head kernel.
// B=2, N=6144, H=256, O=128.
typedef __attribute__((ext_vector_type(2))) float v2f;
typedef __attribute__((ext_vector_type(8))) float v8f;

#define NB 6144
#define HH 256
#define OO 128
#define ROWS (2 * NB)      // 12288 flattened (b,n) rows
#define JC 32              // j-chunk staged in LDS per iteration

__device__ __forceinline__ float lrelu(float x) { return x > 0.0f ? x : 0.01f * x; }
__device__ __forceinline__ float sigmoidf(float x) { return 1.0f / (1.0f + __expf(-x)); }

// ---------------------------------------------------------------------------
// Phase 1: seq_fts[r][o] = sum_h seq[r][h] * W[o][h]   (12288 x 256 x 128)
// One block = 128 rows x 128 cols; 8 waves, each wave = 16 rows x 8 col tiles.
// A (16x4 f32) and B (4x16 f32) operands loaded straight from global as b64.
// ---------------------------------------------------------------------------
__global__ void __launch_bounds__(256) gemm_fts_kernel(
    const float* __restrict__ seq, const float* __restrict__ W,
    float* __restrict__ sf) {
  const int wave  = threadIdx.x >> 5;
  const int lane  = threadIdx.x & 31;
  const int lhalf = lane >> 4;     // 0: K={0,1}, 1: K={2,3}
  const int l15   = lane & 15;
  const int rowA  = blockIdx.x * 128 + wave * 16 + l15;   // A-operand row (M = l15)

  v8f zero = {};
  v8f acc[8];
#pragma unroll
  for (int c = 0; c < 8; ++c) acc[c] = zero;

  const float* arow = seq + rowA * HH + lhalf * 2;
  for (int kk = 0; kk < HH; kk += 4) {
    v2f a = *(const v2f*)(arow + kk);
#pragma unroll
    for (int c = 0; c < 8; ++c) {
      const int o = c * 16 + l15;
      v2f b = *(const v2f*)(W + o * HH + kk + lhalf * 2);
      acc[c] = __builtin_amdgcn_wmma_f32_16x16x4_f32(
          false, a, false, b, (short)0, acc[c], false, false);
    }
  }

  // C layout: VGPR r, lanes 0-15 -> M=r, lanes 16-31 -> M=r+8; N = l15.
  const int ibase = blockIdx.x * 128 + wave * 16 + lhalf * 8;
#pragma unroll
  for (int c = 0; c < 8; ++c) {
    const int o = c * 16 + l15;
#pragma unroll
    for (int r = 0; r < 8; ++r) sf[(ibase + r) * OO + o] = acc[c][r];
  }
}

// ---------------------------------------------------------------------------
// Phase 1b: f1[r] = dot(sf[r,:], w1) + b1 ; f2 likewise.
// ---------------------------------------------------------------------------
__global__ void f12_kernel(const float* __restrict__ sf,
                           const float* __restrict__ w1, const float* __restrict__ b1,
                           const float* __restrict__ w2, const float* __restrict__ b2,
                           float* __restrict__ f1, float* __restrict__ f2) {
  const int r = blockIdx.x * blockDim.x + threadIdx.x;
  if (r >= ROWS) return;
  float a1 = 0.0f, a2 = 0.0f;
  const float* row = sf + r * OO;
#pragma unroll 4
  for (int o = 0; o < OO; ++o) {
    const float v = row[o];
    a1 = fmaf(v, w1[o], a1);
    a2 = fmaf(v, w2[o], a2);
  }
  f1[r] = a1 + b1[0];
  f2[r] = a2 + b2[0];
}

// colsum1[o] = sum_j sf[batch1 row j][o]
__global__ void colsum_kernel(const float* __restrict__ sf, float* __restrict__ cs) {
  const int o = threadIdx.x;   // 128 threads, coalesced across o
  float a = 0.0f;
  const float* p = sf + (size_t)NB * OO + o;
#pragma unroll 4
  for (int j = 0; j < NB; ++j) a += p[j * OO];
  cs[o] = a;
}

// ---------------------------------------------------------------------------
// Phase 2: P = S @ F_b  with S[i][j] = sigmoid(lrelu(f1_0[i]+f2_0[j])
//                                            - lrelu(f1_1[i]+f2_1[j]))
// out[0] = elu(P(F0) + bias); out[1] = elu(colsum1 - P(F1) + bias)
// grid = (48 i-tiles, 2 batches). Block: 256 thr (8 waves).
// Wave w: rows [i0+16w, i0+16w+16) x 128 cols (8 accum tiles).
// F chunk (32 j x 128 o) staged in LDS pre-swizzled into WMMA B-operand
// layout -> each B fetch is a single ds_load_b64.
// ---------------------------------------------------------------------------
__global__ void __launch_bounds__(256) attn_gemm_kernel(
    const float* __restrict__ sf, const float* __restrict__ f1,
    const float* __restrict__ f2, const float* __restrict__ cs,
    const float* __restrict__ bias, float* __restrict__ out) {
  // 8 jg-groups * 8 col tiles * 32 lanes * 2 floats = 4096 floats = 16 KB
  __shared__ float ldsB[(JC / 4) * 8 * 32 * 2];
  __shared__ float ldsF2[2][JC];

  const int t     = threadIdx.x;
  const int wave  = t >> 5;
  const int lane  = t & 31;
  const int lhalf = lane >> 4;
  const int l15   = lane & 15;
  const int i0    = blockIdx.x * 128;
  const int gy    = blockIdx.y;            // batch of the RHS/output
  const int irow  = i0 + wave * 16 + l15;  // A-operand row (M = l15)

  const float a1 = f1[irow];        // f1[0, i]
  const float a2 = f1[NB + irow];   // f1[1, i]

  v8f zero = {};
  v8f acc[8];
#pragma unroll
  for (int c = 0; c < 8; ++c) acc[c] = zero;

  for (int jc = 0; jc < NB; jc += JC) {
    __syncthreads();  // protect LDS from previous iteration's readers
    // Stage F[gy][jc..jc+31][0..127] swizzled: dest float2 index d:
    //   blk = d>>5 = jg*8 + c ; lane = d&31
    //   j = jc + 4*jg + (lane>=16 ? 2 : 0) (+1 for .y) ; o = c*16 + (lane&15)
    for (int d = t; d < 2048; d += 256) {
      const int blk  = d >> 5;
      const int ln   = d & 31;
      const int jg   = blk >> 3;
      const int c    = blk & 7;
      const int jrel = 4 * jg + ((ln >> 4) << 1);
      const int o    = c * 16 + (ln & 15);
      const float* src = sf + ((size_t)gy * NB + jc + jrel) * OO + o;
      ldsB[2 * d]     = src[0];    // K = jrel
      ldsB[2 * d + 1] = src[OO];   // K = jrel + 1
    }
    if (t < JC) {
      ldsF2[0][t] = f2[jc + t];
      ldsF2[1][t] = f2[NB + jc + t];
    }
    __syncthreads();

#pragma unroll
    for (int jg = 0; jg < JC / 4; ++jg) {
      const int jrel = 4 * jg + lhalf * 2;
      // A operand: 2 S values per lane (K = jrel, jrel+1 for row i = irow)
      const float p0a = ldsF2[0][jrel], p0b = ldsF2[0][jrel + 1];
      const float p1a = ldsF2[1][jrel], p1b = ldsF2[1][jrel + 1];
      v2f a;
      a.x = sigmoidf(lrelu(a1 + p0a) - lrelu(a2 + p1a));
      a.y = sigmoidf(lrelu(a1 + p0b) - lrelu(a2 + p1b));
#pragma unroll
      for (int c = 0; c < 8; ++c) {
        const v2f b = *(const v2f*)&ldsB[((jg * 8 + c) * 32 + lane) * 2];
        acc[c] = __builtin_amdgcn_wmma_f32_16x16x4_f32(
            false, a, false, b, (short)0, acc[c], false, false);
      }
    }
  }

  // Epilogue: complement trick for batch 1, bias, ELU.
  const float bs = bias[0];
  const int ibase = i0 + wave * 16 + lhalf * 8;
#pragma unroll
  for (int c = 0; c < 8; ++c) {
    const int o = c * 16 + l15;
    const float csv = (gy == 1) ? cs[o] : 0.0f;
#pragma unroll
    for (int r = 0; r < 8; ++r) {
      float v = acc[c][r];
      if (gy == 1) v = csv - v;
      v += bs;
      out[((size_t)gy * NB + ibase + r) * OO + o] = v > 0.0f ? v : __expf(v) - 1.0f;
    }
  }
}

// ---------------------------------------------------------------------------
extern "C" void kernel_launch(void* const* d_in, const int* in_sizes, int n_in,
                              void* d_out, int out_size, void* d_ws, size_t ws_size,
                              hipStream_t stream) {
  const float* seq  = (const float*)d_in[0];
  const float* Wf   = (const float*)d_in[1];
  const float* w1   = (const float*)d_in[2];
  const float* b1   = (const float*)d_in[3];
  const float* w2   = (const float*)d_in[4];
  const float* b2   = (const float*)d_in[5];
  const float* bias = (const float*)d_in[6];

  float* sf = (float*)d_ws;               // [12288 x 128]
  float* f1 = sf + (size_t)ROWS * OO;     // [12288]
  float* f2 = f1 + ROWS;                  // [12288]
  float* cs = f2 + ROWS;                  // [128]
  float* out = (float*)d_out;             // [2 x 6144 x 128]

  gemm_fts_kernel<<<ROWS / 128, 256, 0, stream>>>(seq, Wf, sf);
  f12_kernel<<<ROWS / 256, 256, 0, stream>>>(sf, w1, b1, w2, b2, f1, f2);
  colsum_kernel<<<1, 128, 0, stream>>>(sf, cs);
  attn_gemm_kernel<<<dim3(NB / 128, 2), 256, 0, stream>>>(sf, f1, f2, cs, bias, out);
}